// KWinnersTakeAll_12223476924647
// MI455X (gfx1250) — compile-verified
//
#include <hip/hip_runtime.h>

// ---------------------------------------------------------------------------
// k-winners-take-all, one workgroup (256 threads = 8 wave32) per row.
// Row is staged to LDS once via CDNA5 GLOBAL_LOAD_ASYNC_TO_LDS_B128,
// threshold found with an exact 4x8-bit LDS radix-select, output written as
// a 0/1 mask with float4 stores.  HBM traffic: 1 read + 1 write per element.
// ---------------------------------------------------------------------------

#ifndef __has_builtin
#define __has_builtin(x) 0
#endif

#if __has_builtin(__builtin_amdgcn_global_load_async_to_lds_b128)
#define HAVE_ASYNC_B128 1
#else
#define HAVE_ASYNC_B128 0
#endif

#if __has_builtin(__builtin_amdgcn_s_wait_asynccnt)
#define HAVE_WAIT_ASYNC 1
#else
#define HAVE_WAIT_ASYNC 0
#endif

#define COLS      4096
#define K_ACTIVE  205            // ceil(0.05 * 4096)
#define TPB       256            // 8 wave32 per workgroup
#define NWAVES    (TPB / 32)
#define HPAD      257            // 257 % 64 == 1 -> per-wave bank stagger

// Builtin signature (from hipcc diagnostic): first param is v4i* in AS(1),
// second is v4i* in AS(3):  (global v4i*, lds v4i*, imm offset, imm cpol)
typedef int v4i __attribute__((ext_vector_type(4)));
typedef __attribute__((address_space(1))) v4i* as1_v4i;
typedef __attribute__((address_space(3))) v4i* as3_v4i;

// Order-preserving monotone map: larger float  <=>  larger unsigned.
__device__ __forceinline__ unsigned ord32(float f) {
  unsigned u = __float_as_uint(f);
  return (u & 0x80000000u) ? ~u : (u | 0x80000000u);
}

__global__ __launch_bounds__(TPB, 4)
void kwta_rowselect(const float* __restrict__ x, float* __restrict__ out) {
  __shared__ float    rowf[COLS];               // 16 KB staged row
  __shared__ unsigned hist[NWAVES][HPAD];       // per-wave private histograms
  __shared__ unsigned sufs[TPB];                // suffix-sum scan buffer
  __shared__ unsigned sh_prefix;
  __shared__ unsigned sh_want;

  const int tid  = (int)threadIdx.x;
  const int wave = tid >> 5;                    // wave32 id, 0..7
  const long long row = (long long)blockIdx.x;
  const float* __restrict__ rowp = x   + row * (long long)COLS;
  float*       __restrict__ outp = out + row * (long long)COLS;

  // ------------------ stage row into LDS (16 B / lane / issue) -------------
#if HAVE_ASYNC_B128
  #pragma unroll
  for (int r = 0; r < COLS / (TPB * 4); ++r) {
    const int e = (r * TPB + tid) * 4;
    __builtin_amdgcn_global_load_async_to_lds_b128(
        (as1_v4i)(rowp + e), (as3_v4i)(&rowf[e]), /*offset=*/0, /*cpol=*/0);
  }
#if HAVE_WAIT_ASYNC
  __builtin_amdgcn_s_wait_asynccnt(0);
#else
  asm volatile("s_wait_asynccnt 0" ::: "memory");
#endif
#else
  #pragma unroll
  for (int r = 0; r < COLS / (TPB * 4); ++r) {
    const int e = (r * TPB + tid) * 4;
    *(float4*)(&rowf[e]) = *(const float4*)(rowp + e);
  }
#endif
  __syncthreads();

  // ------------------ exact radix select of the K-th largest ---------------
  unsigned prefix = 0u;                         // decided high bits of answer
  unsigned want   = K_ACTIVE;                   // rank remaining inside prefix

  #pragma unroll
  for (int pass = 0; pass < 4; ++pass) {
    const int shift = 24 - 8 * pass;
    const unsigned pmask = (pass == 0) ? 0u : (0xFFFFFFFFu << (shift + 8));

    // clear private histograms (each thread owns one bin across all waves)
    #pragma unroll
    for (int h = 0; h < NWAVES; ++h) hist[h][tid] = 0u;
    __syncthreads();

    // count matching elements into this wave's private histogram
    #pragma unroll
    for (int j = 0; j < COLS / TPB; ++j) {
      const unsigned u = ord32(rowf[tid + j * TPB]);
      if ((u & pmask) == prefix) {
        const unsigned bin = (u >> shift) & 0xFFu;
        atomicAdd(&hist[wave][bin], 1u);        // ds_add_u32, bank-staggered
      }
    }
    __syncthreads();

    // reduce private histograms; thread t owns bin t
    unsigned h = 0u;
    #pragma unroll
    for (int w = 0; w < NWAVES; ++w) h += hist[w][tid];
    sufs[tid] = h;
    __syncthreads();

    // suffix sum: sufs[t] = #elements with bin >= t   (Hillis-Steele, 8 steps)
    #pragma unroll
    for (int off = 1; off < TPB; off <<= 1) {
      const unsigned v = (tid + off < TPB) ? sufs[tid + off] : 0u;
      __syncthreads();
      h += v;
      sufs[tid] = h;
      __syncthreads();
    }

    // unique boundary bucket: S[t] >= want  &&  S[t+1] < want
    const unsigned S_t  = sufs[tid];
    const unsigned S_t1 = (tid < TPB - 1) ? sufs[tid + 1] : 0u;
    if (S_t >= want && S_t1 < want) {
      sh_prefix = prefix | ((unsigned)tid << shift);
      sh_want   = want - S_t1;                  // rank inside chosen bucket
    }
    __syncthreads();
    prefix = sh_prefix;
    want   = sh_want;
    __syncthreads();
  }

  // prefix now equals the K-th largest ord32 value exactly.
  const unsigned thr = prefix;

  // ------------------ emit binary mask with b128 stores --------------------
  #pragma unroll
  for (int r = 0; r < COLS / (TPB * 4); ++r) {
    const int e = (r * TPB + tid) * 4;
    const float4 v = *(const float4*)(&rowf[e]);
    float4 m;
    m.x = (ord32(v.x) >= thr) ? 1.0f : 0.0f;
    m.y = (ord32(v.y) >= thr) ? 1.0f : 0.0f;
    m.z = (ord32(v.z) >= thr) ? 1.0f : 0.0f;
    m.w = (ord32(v.w) >= thr) ? 1.0f : 0.0f;
    *(float4*)(outp + e) = m;
  }
}

// ---------------------------------------------------------------------------

extern "C" void kernel_launch(void* const* d_in, const int* in_sizes, int n_in,
                              void* d_out, int out_size, void* d_ws, size_t ws_size,
                              hipStream_t stream) {
  (void)n_in; (void)d_ws; (void)ws_size; (void)out_size;
  const float* x = (const float*)d_in[0];
  float* out = (float*)d_out;
  const int rows = in_sizes[0] / COLS;          // 16384
  kwta_rowselect<<<dim3(rows), dim3(TPB), 0, stream>>>(x, out);
}